// NGCNLayer_32641751450076
// MI455X (gfx1250) — compile-verified
//
#include <hip/hip_runtime.h>

#define IN_FEAT 128
#define OUT_FEAT 128
#define NUM_RELS 16

typedef __attribute__((ext_vector_type(2))) float v2f;
typedef __attribute__((ext_vector_type(8))) float v8f;

// ---------------- kernel 0: zero output + control ints ----------------
__global__ void k_zero(float* __restrict__ out, int n, int* __restrict__ ctrl, int nctrl) {
  int i = blockIdx.x * blockDim.x + threadIdx.x;
  int stride = gridDim.x * blockDim.x;
  for (int j = i; j < n; j += stride) out[j] = 0.0f;
  if (i < nctrl) ctrl[i] = 0;
}

// ---------------- kernel 1: per-relation edge counts ----------------
__global__ void k_count(const int* __restrict__ rel, int ne, int* __restrict__ counts) {
  int i = blockIdx.x * blockDim.x + threadIdx.x;
  int stride = gridDim.x * blockDim.x;
  for (int e = i; e < ne; e += stride) atomicAdd(&counts[rel[e]], 1);
}

// ---------------- kernel 2: tiny exclusive scan (16 bins) ----------------
__global__ void k_scan(const int* __restrict__ counts, int* __restrict__ offsets,
                       int* __restrict__ cursors) {
  if (threadIdx.x == 0 && blockIdx.x == 0) {
    int acc = 0;
    for (int r = 0; r < NUM_RELS; ++r) {
      offsets[r] = acc;
      cursors[r] = acc;
      acc += counts[r];
    }
    offsets[NUM_RELS] = acc;
  }
}

// ---------------- kernel 3: scatter edge ids into relation bins ----------------
__global__ void k_scatter(const int* __restrict__ rel, int ne, int* __restrict__ cursors,
                          int* __restrict__ binned) {
  int i = blockIdx.x * blockDim.x + threadIdx.x;
  int stride = gridDim.x * blockDim.x;
  for (int e = i; e < ne; e += stride) {
    int pos = atomicAdd(&cursors[rel[e]], 1);
    binned[pos] = e;
  }
}

// ---------------- kernel 4: edge-tile GEMM via V_WMMA_F32_16X16X4_F32 ----------------
// Block = 256 threads (8 wave32). blockIdx.y = relation. Each block loops over
// 16-edge tiles of its relation. Per tile: stage 16 gathered h-rows (16x128 f32)
// in LDS (row stride 132 -> conflict-free b64 A-fragment reads), each wave owns
// a 16-column slice of the 128-wide output, B fragments of W_r live in VGPRs for
// the whole block lifetime, 32 WMMAs accumulate C, result scatter-added to out.
#define A_STRIDE 132

__global__ void __launch_bounds__(256)
k_edge_gemm(const float* __restrict__ h, const float* __restrict__ weight,
            const int* __restrict__ src, const int* __restrict__ dst,
            const int* __restrict__ offsets, const int* __restrict__ binned,
            float* __restrict__ out) {
  __shared__ __align__(16) float As[16 * A_STRIDE];
  __shared__ int ssrc[16];
  __shared__ int sdst[16];

  const int r = blockIdx.y;
  const int off = offsets[r];
  const int count = offsets[r + 1] - off;
  const int ntiles = (count + 15) >> 4;

  const int tid = threadIdx.x;
  const int lane = tid & 31;
  const int wave = tid >> 5;   // 0..7 -> output column tile
  const int hi = lane >> 4;    // half-wave: K interleave select
  const int lo = lane & 15;    // M row (A) / N col (B,D)
  const int n0 = wave * 16;

  // Preload this wave's B fragments of W_r into registers: kept across all tiles.
  // B(4x16) layout: lanes0-15 hold K={k,k+1}, lanes16-31 hold K={k+2,k+3}, N=lane%16.
  const float* Wr = weight + (size_t)r * IN_FEAT * OUT_FEAT;
  v2f bfrag[32];
#pragma unroll
  for (int kk = 0; kk < 32; ++kk) {
    int k = kk * 4 + 2 * hi;
    v2f b;
    b.x = Wr[(size_t)k * OUT_FEAT + n0 + lo];
    b.y = Wr[(size_t)(k + 1) * OUT_FEAT + n0 + lo];
    bfrag[kk] = b;
  }

  for (int t = blockIdx.x; t < ntiles; t += gridDim.x) {
    // --- stage edge metadata ---
    if (tid < 16) {
      int idx = t * 16 + tid;
      int e = (idx < count) ? binned[off + idx] : -1;
      ssrc[tid] = (e >= 0) ? src[e] : -1;
      sdst[tid] = (e >= 0) ? dst[e] : -1;
    }
    __syncthreads();

    // --- gather A tile: 16 rows of h (L2-resident) into LDS ---
    {
      int row = tid >> 4;        // 0..15
      int c4 = (tid & 15) * 4;   // two float4 segments per thread
      int s = ssrc[row];
      float4 v0, v1;
      if (s >= 0) {
        const float* hp = h + (size_t)s * IN_FEAT;
        v0 = *(const float4*)(hp + c4);
        v1 = *(const float4*)(hp + c4 + 64);
      } else {
        v0 = make_float4(0.f, 0.f, 0.f, 0.f);
        v1 = v0;
      }
      *(float4*)&As[row * A_STRIDE + c4] = v0;
      *(float4*)&As[row * A_STRIDE + c4 + 64] = v1;
    }
    __syncthreads();

    // --- 16x128 = (16x128) x (128x16-per-wave), K in steps of 4 ---
    v8f acc = {};
#pragma unroll
    for (int kk = 0; kk < 32; ++kk) {
      int k = kk * 4 + 2 * hi;
      // A(16x4): lane holds {A[m][k], A[m][k+1]} (half-wave K interleave)
      v2f a = *(const v2f*)&As[lo * A_STRIDE + k];
      acc = __builtin_amdgcn_wmma_f32_16x16x4_f32(
          /*neg_a=*/false, a, /*neg_b=*/false, bfrag[kk],
          /*c_mod=*/(short)0, acc, /*reuse_a=*/false, /*reuse_b=*/false);
    }

    // --- scatter-add D(16x16) to out[dst]: VGPR j holds M=j (lanes 0-15) / M=j+8 ---
#pragma unroll
    for (int j = 0; j < 8; ++j) {
      int row = j + 8 * hi;
      int d = sdst[row];
      if (d >= 0) atomicAdd(&out[(size_t)d * OUT_FEAT + n0 + lo], acc[j]);
    }
    __syncthreads();  // As/sdst reused next tile
  }
}

// ---------------- kernel 5: out = relu(out + bias), in place ----------------
__global__ void k_bias_relu(float* __restrict__ out, const float* __restrict__ bias, int n) {
  int i = blockIdx.x * blockDim.x + threadIdx.x;
  int stride = gridDim.x * blockDim.x;
  for (int j = i; j < n; j += stride) {
    float v = out[j] + bias[j & (OUT_FEAT - 1)];
    out[j] = v > 0.f ? v : 0.f;
  }
}

extern "C" void kernel_launch(void* const* d_in, const int* in_sizes, int n_in,
                              void* d_out, int out_size, void* d_ws, size_t ws_size,
                              hipStream_t stream) {
  const float* h      = (const float*)d_in[0];
  const float* weight = (const float*)d_in[1];
  const float* bias   = (const float*)d_in[2];
  const int*   src    = (const int*)d_in[3];
  const int*   dst    = (const int*)d_in[4];
  const int*   rel    = (const int*)d_in[5];

  float* out = (float*)d_out;
  const int ne = in_sizes[3];
  const int nout = out_size;  // N_NODES * OUT_FEAT

  // workspace layout
  int* counts  = (int*)d_ws;        // 16
  int* offsets = counts + 16;       // 17
  int* cursors = offsets + 17;      // 16
  int* binned  = cursors + 16;      // ne
  const int nctrl = 16 + 17 + 16;

  k_zero<<<(nout + 255) / 256, 256, 0, stream>>>(out, nout, counts, nctrl);
  k_count<<<2048, 256, 0, stream>>>(rel, ne, counts);
  k_scan<<<1, 32, 0, stream>>>(counts, offsets, cursors);
  k_scatter<<<2048, 256, 0, stream>>>(rel, ne, cursors, binned);

  dim3 grid(128, NUM_RELS);
  k_edge_gemm<<<grid, 256, 0, stream>>>(h, weight, src, dst, offsets, binned, out);

  k_bias_relu<<<(nout + 255) / 256, 256, 0, stream>>>(out, bias, nout);
}